// PhysicsPriorGenerator_26603027431638
// MI455X (gfx1250) — compile-verified
//
#include <hip/hip_runtime.h>
#include <hip/hip_bf16.h>

typedef __attribute__((ext_vector_type(2))) float v2f;
typedef __attribute__((ext_vector_type(8))) float v8f;

#define D_DEPTH 512
#define A_ANG   256
#define MAXD    50.0f

// One block = one (tensor, b) pair; 256 threads = 8 waves, wave w owns angle
// columns [32w, 32w+32). Pass 1: per-wave WMMA depth-sums for two 16-column
// tiles (ones-matrix trick -> sums replicated across all C rows, so lane L
// reads its column's sum directly from acc[0]). Pass 2: recompute exp and
// stream normalized values with non-temporal stores (output >> L2).
__global__ __launch_bounds__(256) void prior_kernel(const float* __restrict__ p,
                                                    const float* __restrict__ pc,
                                                    float* __restrict__ out)
{
    const int blk    = blockIdx.x;        // 0..511
    const int tensor = blk >> 8;          // 0 -> p, 1 -> p_calib
    const int b      = blk & 255;
    const float* prm = tensor ? pc : p;
    const float d       = prm[b * 3 + 1];
    const float theta_p = prm[b * 3 + 2];

    const int tid  = threadIdx.x;
    const int lane = tid & 31;
    const int wave = tid >> 5;
    const int h    = lane >> 4;           // half-wave: which K pair we feed
    const int m    = lane & 15;           // column-in-tile

    const int aA    = wave * 32 + m;      // tile-lo column this lane feeds
    const int aB    = aA + 16;            // tile-hi column this lane feeds
    const int a_own = wave * 32 + lane;   // column this lane stores in pass 2

    const float ANG_STEP   = 60.0f / 255.0f;     // linspace(-30,30,256)
    const float DEPTH_STEP = 50.0f / 511.0f;     // linspace(0,50,512)
    const float DEG2RAD    = 0.017453292519943295f;
    const float EXPC       = -2.0f * 1.4426950408889634f; // -(1/(2*0.25))*log2(e)

    const float angA = (theta_p + (-30.0f + (float)aA * ANG_STEP)) * DEG2RAD;
    const float angB = (theta_p + (-30.0f + (float)aB * ANG_STEP)) * DEG2RAD;
    const float dthA = d / __cosf(angA);
    const float dthB = d / __cosf(angB);
    const float mskA = (dthA > 0.0f && dthA < MAXD) ? 1.0f : 0.0f;
    const float mskB = (dthB > 0.0f && dthB < MAXD) ? 1.0f : 0.0f;

    v2f ones = {1.0f, 1.0f};              // A matrix (16x4 all-ones)
    v8f accA = {};                        // C/D tile-lo (columns aA)
    v8f accB = {};                        // C/D tile-hi (columns aB)

    // ---- Pass 1: depth sums via V_WMMA_F32_16X16X4_F32 ----
    // Lane supplies B(4x16) slots for k = 4c + 2h + {0,1}, column lane%16.
    // Sum over the chunk is invariant to B's exact K-slot striping.
    float z = (float)(2 * h) * DEPTH_STEP;
    const float z4 = 4.0f * DEPTH_STEP;
    #pragma unroll 2
    for (int c = 0; c < 128; ++c) {
        const float z0 = z, z1 = z + DEPTH_STEP;
        const float dA0 = z0 - dthA, dA1 = z1 - dthA;
        const float dB0 = z0 - dthB, dB1 = z1 - dthB;
        v2f bA, bB;
        bA[0] = __builtin_amdgcn_exp2f(EXPC * dA0 * dA0) * mskA;
        bA[1] = __builtin_amdgcn_exp2f(EXPC * dA1 * dA1) * mskA;
        bB[0] = __builtin_amdgcn_exp2f(EXPC * dB0 * dB0) * mskB;
        bB[1] = __builtin_amdgcn_exp2f(EXPC * dB1 * dB1) * mskB;
        // D = A(16x4 ones) x B(4x16 exp) + C : every C row = column sums
        accA = __builtin_amdgcn_wmma_f32_16x16x4_f32(
                   false, ones, false, bA, (short)0, accA, false, false);
        accB = __builtin_amdgcn_wmma_f32_16x16x4_f32(
                   false, ones, false, bB, (short)0, accB, false, false);
        z += z4;
    }

    // acc[0] at lane L = depth-sum of column (L%16) of that tile (rows identical).
    const float sum     = (h == 0) ? accA[0] : accB[0];   // sum for column a_own
    const float denom   = (sum > 0.0f) ? sum : 1.0f;
    const float dth_own = (h == 0) ? dthA : dthB;
    const float msk_own = (h == 0) ? mskA : mskB;
    const float scale   = msk_own / denom;

    // ---- Pass 2: recompute exp, normalize, stream out (non-temporal) ----
    // Wave writes 32 consecutive 'a' per k -> 128B contiguous per store issue.
    float* dst = out + (size_t)blk * (size_t)(D_DEPTH * A_ANG) + a_own;
    float zz = 0.0f;
    #pragma unroll 4
    for (int k = 0; k < D_DEPTH; ++k) {
        const float df  = zz - dth_own;
        const float val = __builtin_amdgcn_exp2f(EXPC * df * df) * scale;
        __builtin_nontemporal_store(val, dst);
        dst += A_ANG;
        zz  += DEPTH_STEP;
    }
}

extern "C" void kernel_launch(void* const* d_in, const int* in_sizes, int n_in,
                              void* d_out, int out_size, void* d_ws, size_t ws_size,
                              hipStream_t stream) {
    const float* p  = (const float*)d_in[0];   // (256,3) fp32
    const float* pc = (const float*)d_in[1];   // (256,3) fp32
    float* out = (float*)d_out;                // 2 x 256 x 512 x 256 fp32
    hipLaunchKernelGGL(prior_kernel, dim3(2 * 256), dim3(256), 0, stream,
                       p, pc, out);
}